// LangRNNExtractor_10385230921843
// MI455X (gfx1250) — compile-verified
//
#include <hip/hip_runtime.h>

#define VOCAB 128000
#define SEQ   64
#define HID   128
#define G3    384          // 3*HID
#define OUTD  64
#define BATCH 256
#define EXTRA 128
#define OBSW  (EXTRA + SEQ)   // 192 (also the output row width: 128 + 64)

#define HPAD  132          // padded LDS row stride for h   (breaks stride-128 bank pattern)
#define GPAD  388          // padded LDS row stride for gates
#define NTHR  768          // 24 waves per workgroup

typedef __attribute__((ext_vector_type(2))) float v2f;
typedef __attribute__((ext_vector_type(8))) float v8f;

// ---------------------------------------------------------------------------
// Kernel 1: embedding gather.  xp[b][s][j] = w_ih[j][id(b,s)] + b_ih[j]
// Pure bandwidth: every gathered element is on its own cacheline; NT loads.
// ---------------------------------------------------------------------------
__global__ __launch_bounds__(256) void lang_gather_kernel(
    const float* __restrict__ obs, const float* __restrict__ w_ih,
    const float* __restrict__ b_ih, float* __restrict__ xp)
{
  long tid = (long)blockIdx.x * blockDim.x + threadIdx.x;
  const long total = (long)BATCH * SEQ * G3;
  if (tid >= total) return;
  int j  = (int)(tid % G3);
  int bs = (int)(tid / G3);
  int b  = bs >> 6;          // / SEQ
  int s  = bs & (SEQ - 1);
  int id = (int)obs[(long)b * OBSW + EXTRA + s];
  if (id < 0) id = 0;
  if (id >= VOCAB) id = VOCAB - 1;
  float w = __builtin_nontemporal_load(&w_ih[(long)j * VOCAB + id]);
  xp[(long)bs * G3 + j] = w + b_ih[j];
}

// ---------------------------------------------------------------------------
// Kernel 2: GRU over 64 steps + FC head, f32 WMMA 16x16x4.
// One workgroup = 24 waves = 768 threads, owns 16 batch rows.
// Wave w owns gate N-tile [16w, 16w+16); its w_hh B-fragments live in VGPRs.
// Dual accumulator chains halve the per-step WMMA dependency depth; x_proj
// values are prefetched via pure global loads (integer offsets, no generic
// pointers -> no flat/DScnt interference with the A-fragment ds pipeline).
// ---------------------------------------------------------------------------
__global__ __launch_bounds__(NTHR) void lang_gru_kernel(
    const float* __restrict__ obs, const float* __restrict__ w_hh,
    const float* __restrict__ b_hh, const float* __restrict__ fc_w,
    const float* __restrict__ fc_b, const float* __restrict__ xp,
    float* __restrict__ out)
{
  __shared__ float sh_h[16 * HPAD];   // h tile   [16][128] (padded)
  __shared__ float sh_g[16 * GPAD];   // gh tile  [16][384] (padded); reused for FC

  const int tid   = threadIdx.x;
  const int wave  = tid >> 5;
  const int lane  = tid & 31;
  const int ln16  = lane & 15;
  const int half  = lane >> 4;        // 0: lanes 0-15, 1: lanes 16-31
  const int bbase = blockIdx.x * 16;

  // pass-through copy of the non-language part of the observation
  for (int i = tid; i < 16 * EXTRA; i += NTHR) {
    int m = i >> 7, e = i & 127;
    out[(long)(bbase + m) * OBSW + e] = obs[(long)(bbase + m) * OBSW + e];
  }
  // h0 = 0
  for (int i = tid; i < 16 * HPAD; i += NTHR) sh_h[i] = 0.0f;

  // Preload B fragments: B[k][n] = w_hh[n][k].
  // Lane layout (mirror of A 16x4): n = 16*wave + ln16, k = 4*ks + 2*half + {0,1}
  v2f bw[32];
  {
    const float* wrow = w_hh + (long)(16 * wave + ln16) * HID + 2 * half;
    #pragma unroll
    for (int ks = 0; ks < 32; ++ks) {
      bw[ks].x = wrow[4 * ks];
      bw[ks].y = wrow[4 * ks + 1];
    }
  }

  // Step-invariant elementwise coordinates for this thread (<=3 elements),
  // b_hh hoisted into registers; x_proj addressed by integer offsets from the
  // global-AS kernel argument (keeps loads on the global path).
  const long xbase = (long)bbase * SEQ * G3;
  long xoff[3];
  float br[3], bz[3], bn[3];
  #pragma unroll
  for (int t = 0; t < 3; ++t) {
    int i = tid + NTHR * t;
    if (i < 16 * HID) {
      int m = i >> 7, k = i & 127;
      br[t] = b_hh[k];
      bz[t] = b_hh[HID + k];
      bn[t] = b_hh[2 * HID + k];
      xoff[t] = xbase + (long)m * SEQ * G3 + k;   // advances by G3 per step
    }
  }
  __syncthreads();

  for (int s = 0; s < SEQ; ++s) {
    // ---- prefetch this step's x_proj values (hides behind the WMMA chain) --
    float ir[3], iz[3], inn[3];
    #pragma unroll
    for (int t = 0; t < 3; ++t) {
      if (tid + NTHR * t < 16 * HID) {
        ir[t]  = xp[xoff[t]];
        iz[t]  = xp[xoff[t] + HID];
        inn[t] = xp[xoff[t] + 2 * HID];
        xoff[t] += G3;
      }
    }

    // ---- gh = h @ w_hh.T : two interleaved v_wmma_f32_16x16x4_f32 chains ---
    v8f c0 = {}, c1 = {};
    #pragma unroll
    for (int ks = 0; ks < 32; ks += 2) {
      v2f a0 = *(const v2f*)&sh_h[ln16 * HPAD + 4 * ks + 2 * half];
      v2f a1 = *(const v2f*)&sh_h[ln16 * HPAD + 4 * (ks + 1) + 2 * half];
      c0 = __builtin_amdgcn_wmma_f32_16x16x4_f32(false, a0, false, bw[ks],
                                                 (short)0, c0, false, false);
      c1 = __builtin_amdgcn_wmma_f32_16x16x4_f32(false, a1, false, bw[ks + 1],
                                                 (short)0, c1, false, false);
    }
    // D layout: vgpr j -> row (j + 8*half), col = 16*wave + ln16
    #pragma unroll
    for (int j = 0; j < 8; ++j)
      sh_g[(j + 8 * half) * GPAD + 16 * wave + ln16] = c0[j] + c1[j];
    __syncthreads();

    // ---- elementwise gate update (in-place on sh_h) ----
    #pragma unroll
    for (int t = 0; t < 3; ++t) {
      int i = tid + NTHR * t;
      if (i < 16 * HID) {
        int m = i >> 7, k = i & 127;
        float gr = sh_g[m * GPAD + k]           + br[t];
        float gz = sh_g[m * GPAD + HID + k]     + bz[t];
        float gn = sh_g[m * GPAD + 2 * HID + k] + bn[t];
        float r = 1.0f / (1.0f + __expf(-(ir[t] + gr)));
        float z = 1.0f / (1.0f + __expf(-(iz[t] + gz)));
        float n = tanhf(inn[t] + r * gn);
        float hp = sh_h[m * HPAD + k];
        sh_h[m * HPAD + k] = (1.0f - z) * n + z * hp;
      }
    }
    __syncthreads();
  }

  // ---- FC head: out = h @ fc_w.T + fc_b (waves 0..3, WMMA again) ----
  if (wave < 4) {
    v2f bf[32];
    const float* frow = fc_w + (long)(16 * wave + ln16) * HID + 2 * half;
    #pragma unroll
    for (int ks = 0; ks < 32; ++ks) {
      bf[ks].x = frow[4 * ks];
      bf[ks].y = frow[4 * ks + 1];
    }
    v8f c0 = {}, c1 = {};
    #pragma unroll
    for (int ks = 0; ks < 32; ks += 2) {
      v2f a0 = *(const v2f*)&sh_h[ln16 * HPAD + 4 * ks + 2 * half];
      v2f a1 = *(const v2f*)&sh_h[ln16 * HPAD + 4 * (ks + 1) + 2 * half];
      c0 = __builtin_amdgcn_wmma_f32_16x16x4_f32(false, a0, false, bf[ks],
                                                 (short)0, c0, false, false);
      c1 = __builtin_amdgcn_wmma_f32_16x16x4_f32(false, a1, false, bf[ks + 1],
                                                 (short)0, c1, false, false);
    }
    #pragma unroll
    for (int j = 0; j < 8; ++j)
      sh_g[(j + 8 * half) * GPAD + 16 * wave + ln16] = c0[j] + c1[j];
  }
  __syncthreads();

  for (int i = tid; i < 16 * OUTD; i += NTHR) {
    int m = i >> 6, o = i & 63;
    out[(long)(bbase + m) * OBSW + EXTRA + o] = sh_g[m * GPAD + o] + fc_b[o];
  }
}

// ---------------------------------------------------------------------------
extern "C" void kernel_launch(void* const* d_in, const int* in_sizes, int n_in,
                              void* d_out, int out_size, void* d_ws, size_t ws_size,
                              hipStream_t stream) {
  (void)in_sizes; (void)n_in; (void)out_size; (void)ws_size;
  const float* obs  = (const float*)d_in[0];
  const float* w_ih = (const float*)d_in[1];
  const float* w_hh = (const float*)d_in[2];
  const float* b_ih = (const float*)d_in[3];
  const float* b_hh = (const float*)d_in[4];
  const float* fc_w = (const float*)d_in[5];
  const float* fc_b = (const float*)d_in[6];
  float* out = (float*)d_out;
  float* xp  = (float*)d_ws;   // BATCH*SEQ*G3 floats = 25.2 MB of scratch

  const long total = (long)BATCH * SEQ * G3;              // 6,291,456
  const int  gblocks = (int)((total + 255) / 256);        // 24,576
  lang_gather_kernel<<<gblocks, 256, 0, stream>>>(obs, w_ih, b_ih, xp);
  lang_gru_kernel<<<BATCH / 16, NTHR, 0, stream>>>(obs, w_hh, b_hh,
                                                   fc_w, fc_b, xp, out);
}